// FNNStack_v3_63350767616474
// MI455X (gfx1250) — compile-verified
//
#include <hip/hip_runtime.h>
#include <math.h>

typedef __attribute__((ext_vector_type(2))) float v2f;
typedef __attribute__((ext_vector_type(8))) float v8f;

#define D_DIM 128

// -------------------- small utility kernels --------------------

__global__ void zero_kernel(float* __restrict__ p, int n) {
    int i = blockIdx.x * blockDim.x + threadIdx.x;
    if (i < n) p[i] = 0.0f;
}

// histogram of dst node ids (counts as float; E=2^19 exactly representable)
__global__ void hist_kernel(const int* __restrict__ dst, float* __restrict__ cnt, int E) {
    int e = blockIdx.x * blockDim.x + threadIdx.x;
    if (e < E) atomicAdd(&cnt[dst[e]], 1.0f);
}

// t_accum[d] = sum_n cnt[n] * emb_t[n][d]   (128-wide, coalesced per node row)
__global__ void taccum_kernel(const float* __restrict__ emb_t, const float* __restrict__ cnt,
                              float* __restrict__ tacc, int NT) {
    int d = threadIdx.x; // blockDim.x == 128
    float partial = 0.0f;
    for (int n = blockIdx.x; n < NT; n += gridDim.x)
        partial += cnt[n] * emb_t[(size_t)n * D_DIM + d];
    atomicAdd(&tacc[d], partial);
}

// v = (WQ^T @ (WK @ t_accum)) / sqrt(128)   -- two tiny 128x128 matvecs, 1 block
__global__ void matvec_kernel(const float* __restrict__ WK, const float* __restrict__ WQ,
                              const float* __restrict__ tacc, float* __restrict__ v) {
    __shared__ float t[D_DIM];
    __shared__ float ks[D_DIM];
    int i = threadIdx.x; // 0..127
    t[i] = tacc[i];
    __syncthreads();
    float s = 0.0f;
    for (int j = 0; j < D_DIM; ++j) s += WK[(size_t)i * D_DIM + j] * t[j];
    ks[i] = s;
    __syncthreads();
    float u = 0.0f;
    for (int r = 0; r < D_DIM; ++r) u += WQ[(size_t)r * D_DIM + i] * ks[r];
    v[i] = u * 0.08838834764831845f; // 1/sqrt(128)
}

// sdot[n] = relu(emb_s[n] . v)  -- one wave32 per node, float4 coalesced loads
__global__ void sdot_kernel(const float* __restrict__ emb_s, const float* __restrict__ v,
                            float* __restrict__ sdot, int NS) {
    int lane = threadIdx.x & 31;
    int node = (blockIdx.x * blockDim.x + threadIdx.x) >> 5;
    if (node >= NS) return;
    const float4* row = (const float4*)(emb_s + (size_t)node * D_DIM);
    const float4* vv  = (const float4*)v;
    float4 a = row[lane], b = vv[lane];
    float dot = a.x * b.x + a.y * b.y + a.z * b.z + a.w * b.w;
    #pragma unroll
    for (int m = 16; m >= 1; m >>= 1) dot += __shfl_xor(dot, m);
    if (lane == 0) sdot[node] = fmaxf(dot, 0.0f);
}

// build initial y: 6-feature L2 normalize + linear + relu; cache x_s_d/x_s_c per edge
__global__ void yinit_kernel(const int* __restrict__ src, const int* __restrict__ dstp,
                             const float* __restrict__ sdot,
                             const float* __restrict__ x_s, const float* __restrict__ x_t,
                             const float* __restrict__ at_s, const float* __restrict__ at_t,
                             const float* __restrict__ ew, const float* __restrict__ Wl,
                             const float* __restrict__ bl,
                             float* __restrict__ y, float* __restrict__ xc, int E) {
    int e = blockIdx.x * blockDim.x + threadIdx.x;
    if (e >= E) return;
    int s = src[e], d = dstp[e];
    float a   = sdot[s];
    float xsd = x_s[(size_t)s * 4 + 1];
    float xsc = x_t[(size_t)d * 4 + 1];
    const float4* ps = (const float4*)(at_s + ((size_t)E + e) * 8); // at_s[1,e,:]
    float4 h0 = ps[0], h1 = ps[1];
    float as = (h0.x + h0.y + h0.z + h0.w + h1.x + h1.y + h1.z + h1.w) * 0.125f;
    const float4* pt = (const float4*)(at_t + ((size_t)E + e) * 8);
    float4 q0 = pt[0], q1 = pt[1];
    float at = (q0.x + q0.y + q0.z + q0.w + q1.x + q1.y + q1.z + q1.w) * 0.125f;
    float w = ew[e];
    float n2 = a * a + xsd * xsd + xsc * xsc + as * as + at * at + w * w;
    float inv = 1.0f / fmaxf(sqrtf(n2), 1e-12f);
    float yv = (a * Wl[0] + xsd * Wl[1] + xsc * Wl[2] + as * Wl[3] + at * Wl[4] + w * Wl[5]) * inv
               + bl[0];
    y[e] = fmaxf(yv, 0.0f);
    xc[(size_t)e * 2]     = xsd;
    xc[(size_t)e * 2 + 1] = xsc;
}

// segment sums by src and dst via float atomics
__global__ void scatter_kernel(const int* __restrict__ src, const int* __restrict__ dstp,
                               const float* __restrict__ y,
                               float* __restrict__ Ssrc, float* __restrict__ Sdst, int E) {
    int e = blockIdx.x * blockDim.x + threadIdx.x;
    if (e >= E) return;
    float yv = y[e];
    atomicAdd(&Ssrc[src[e]], yv);
    atomicAdd(&Sdst[dstp[e]], yv);
}

// -------------------- WMMA per-edge MLP --------------------
// Wave handles 16 edges: coef[16x5] @ W1^T[5x64] via V_WMMA_F32_16X16X4_F32
// (K split 4+1, N split into 4 tiles of 16), then LayerNorm(64) via half-wave
// shfl_xor butterflies, 64->1 dot, y *= relu(out).
__global__ void mlp_wmma_kernel(const int* __restrict__ src, const int* __restrict__ dstp,
                                const float* __restrict__ Ssrc, const float* __restrict__ Sdst,
                                const float* __restrict__ xc,
                                const float* __restrict__ W1, const float* __restrict__ b1,
                                const float* __restrict__ g1, const float* __restrict__ be1,
                                const float* __restrict__ W2, const float* __restrict__ b2,
                                float* __restrict__ y, int E) {
    int lane = threadIdx.x & 31;
    int wave = (blockIdx.x * blockDim.x + threadIdx.x) >> 5;
    int base = wave * 16;
    if (base >= E) return; // wave-uniform; executing waves keep EXEC all-ones
    int col = lane & 15;
    bool hi = lane >= 16;

    // B tiles (4x16 f32 each): VGPR0 = row K=kb, VGPR1 = row K=kb+1,
    // kb = 0 for lanes 0-15, 2 for lanes 16-31 (mirrors f32 A layout).
    v2f B1t[4], B2t[4];
    float b1v[4], g1v[4], bev[4], w2v[4];
    #pragma unroll
    for (int t = 0; t < 4; ++t) {
        int c = t * 16 + col;
        int kb = hi ? 2 : 0;
        B1t[t].x = W1[c * 5 + kb];
        B1t[t].y = W1[c * 5 + kb + 1];
        B2t[t].x = hi ? 0.0f : W1[c * 5 + 4]; // K=4 row; K=5..7 are zero padding
        B2t[t].y = 0.0f;
        b1v[t] = b1[c]; g1v[t] = g1[c]; bev[t] = be1[c]; w2v[t] = W2[c];
    }

    // Each lane loads the full coef row of edge (base + col); lanes 16-31
    // duplicate the loads (L2-hot) so no shuffles are needed to build A.
    int e = base + col;
    int s = src[e], d = dstp[e];
    float f0 = y[e];
    float f1 = Ssrc[s];
    float f2 = xc[(size_t)e * 2];
    float f3 = Sdst[d];
    float f4 = xc[(size_t)e * 2 + 1];
    float inv = 1.0f / fmaxf(sqrtf(f0 * f0 + f1 * f1 + f2 * f2 + f3 * f3 + f4 * f4), 1e-12f);
    f0 *= inv; f1 *= inv; f2 *= inv; f3 *= inv; f4 *= inv;

    // A (16x4 f32): lane=M row; VGPR0/1 hold K0,K1 (low lanes) / K2,K3 (high lanes)
    v2f A1, A2;
    A1.x = hi ? f2 : f0;
    A1.y = hi ? f3 : f1;
    A2.x = hi ? 0.0f : f4;
    A2.y = 0.0f;

    float h[4][8];
    #pragma unroll
    for (int t = 0; t < 4; ++t) {
        v8f acc = {0.f, 0.f, 0.f, 0.f, 0.f, 0.f, 0.f, 0.f};
        acc = __builtin_amdgcn_wmma_f32_16x16x4_f32(false, A1, false, B1t[t],
                                                    (short)0, acc, false, false);
        acc = __builtin_amdgcn_wmma_f32_16x16x4_f32(false, A2, false, B2t[t],
                                                    (short)0, acc, false, false);
        #pragma unroll
        for (int r = 0; r < 8; ++r) h[t][r] = fmaxf(acc[r] + b1v[t], 0.0f);
    }

    // LayerNorm stats per row (row M=r for lanes 0-15, M=r+8 for lanes 16-31):
    // reduce 4 tile values per lane across the 16-lane half (masks 1,2,4,8).
    float sum[8], sq[8];
    #pragma unroll
    for (int r = 0; r < 8; ++r) {
        float s1 = h[0][r] + h[1][r] + h[2][r] + h[3][r];
        float s2 = h[0][r] * h[0][r] + h[1][r] * h[1][r]
                 + h[2][r] * h[2][r] + h[3][r] * h[3][r];
        #pragma unroll
        for (int m = 8; m >= 1; m >>= 1) { s1 += __shfl_xor(s1, m); s2 += __shfl_xor(s2, m); }
        sum[r] = s1; sq[r] = s2;
    }

    float cacc[8];
    #pragma unroll
    for (int r = 0; r < 8; ++r) {
        float mean = sum[r] * (1.0f / 64.0f);
        float var  = sq[r] * (1.0f / 64.0f) - mean * mean;
        float rstd = rsqrtf(var + 1e-5f);
        float cs = 0.0f;
        #pragma unroll
        for (int t = 0; t < 4; ++t) {
            float hn = (h[t][r] - mean) * rstd * g1v[t] + bev[t];
            cs += hn * w2v[t];
        }
        cacc[r] = cs;
    }
    #pragma unroll
    for (int r = 0; r < 8; ++r) {
        #pragma unroll
        for (int m = 8; m >= 1; m >>= 1) cacc[r] += __shfl_xor(cacc[r], m);
    }

    // lane 0 writes rows 0..7, lane 16 writes rows 8..15 (this wave owns them)
    if (col == 0) {
        int rbase = hi ? 8 : 0;
        float bias2 = b2[0];
        #pragma unroll
        for (int r = 0; r < 8; ++r) {
            float coef = fmaxf(cacc[r] + bias2, 0.0f);
            y[base + rbase + r] = y[base + rbase + r] * coef;
        }
    }
}

// -------------------- driver --------------------

extern "C" void kernel_launch(void* const* d_in, const int* in_sizes, int n_in,
                              void* d_out, int out_size, void* d_ws, size_t ws_size,
                              hipStream_t stream) {
    const float* emb_s = (const float*)d_in[0];
    const float* emb_t = (const float*)d_in[1];
    const float* at_s  = (const float*)d_in[2];
    const float* at_t  = (const float*)d_in[3];
    const float* x_s   = (const float*)d_in[4];
    const float* x_t   = (const float*)d_in[5];
    const int*   eidx  = (const int*)d_in[6];
    const float* ew    = (const float*)d_in[7];
    const float* WQ    = (const float*)d_in[8];
    const float* WK    = (const float*)d_in[9];
    const float* Wl    = (const float*)d_in[10];
    const float* bl    = (const float*)d_in[11];
    const float* W1    = (const float*)d_in[12];
    const float* b1    = (const float*)d_in[13];
    const float* g1    = (const float*)d_in[14];
    const float* be1   = (const float*)d_in[15];
    const float* W2    = (const float*)d_in[16];
    const float* b2    = (const float*)d_in[17];

    int NS = in_sizes[0] / 128;
    int NT = in_sizes[1] / 128;
    int E  = in_sizes[7]; // edge_weight has E elements

    const int* src  = eidx;
    const int* dstp = eidx + E;

    float* y  = (float*)d_out;
    float* ws = (float*)d_ws;
    float* cnt  = ws;
    float* tacc = cnt + NT;
    float* v    = tacc + 128;
    float* sdot = v + 128;
    float* Ssrc = sdot + NS;
    float* Sdst = Ssrc + NS;   // adjacent to Ssrc -> zeroed together
    float* xc   = Sdst + NT;

    zero_kernel<<<(NT + 128 + 255) / 256, 256, 0, stream>>>(cnt, NT + 128);
    hist_kernel<<<(E + 255) / 256, 256, 0, stream>>>(dstp, cnt, E);
    taccum_kernel<<<256, 128, 0, stream>>>(emb_t, cnt, tacc, NT);
    matvec_kernel<<<1, 128, 0, stream>>>(WK, WQ, tacc, v);
    sdot_kernel<<<(NS * 32 + 255) / 256, 256, 0, stream>>>(emb_s, v, sdot, NS);
    yinit_kernel<<<(E + 255) / 256, 256, 0, stream>>>(src, dstp, sdot, x_s, x_t,
                                                      at_s, at_t, ew, Wl, bl, y, xc, E);
    for (int it = 0; it < 3; ++it) {
        zero_kernel<<<(NS + NT + 255) / 256, 256, 0, stream>>>(Ssrc, NS + NT);
        scatter_kernel<<<(E + 255) / 256, 256, 0, stream>>>(src, dstp, y, Ssrc, Sdst, E);
        int waves  = E / 16;            // E = 2^19, divides exactly
        int blocks = (waves + 7) / 8;   // 8 wave32 per 256-thread block
        mlp_wmma_kernel<<<blocks, 256, 0, stream>>>(src, dstp, Ssrc, Sdst, xc,
                                                    W1, b1, g1, be1, W2, b2, y, E);
    }
}